// SoftSkeletonize_87591563035237
// MI455X (gfx1250) — compile-verified
//
#include <hip/hip_runtime.h>
#include <stdint.h>

// soft_skel (morphological skeletonization), 41 fused scan steps.
// No matmul content -> WMMA not applicable; CDNA5 feature used instead:
// Tensor Data Mover 2D tile DMA (tensor_load_to_lds) + s_wait_tensorcnt
// to stage halo'd stencil tiles into LDS.

#define TILE 32
#define HALO 3
#define LW (TILE + 2 * HALO)   // 38: input tile width (halo 3)
#define EW (TILE + 4)          // 36: erode-1 region (halo 2)
#define DW (TILE + 2)          // 34: erode-2 region (halo 1)

typedef unsigned int u32x4 __attribute__((ext_vector_type(4)));
typedef int          i32x4 __attribute__((ext_vector_type(4)));
typedef int          i32x8 __attribute__((ext_vector_type(8)));

#if defined(__HIP_DEVICE_COMPILE__) && __has_builtin(__builtin_amdgcn_tensor_load_to_lds)
#define HAVE_TDM 1
#else
#define HAVE_TDM 0
#endif

#if HAVE_TDM
// One TDM descriptor: load a tile_h x tile_w fp32 tile (row stride
// row_stride_elems in global) contiguously into LDS at lds_off.
__device__ __forceinline__ void tdm_load_2d(unsigned lds_off, const void* gptr,
                                            unsigned tile_w, unsigned tile_h,
                                            unsigned row_stride_elems) {
  unsigned long long ga = (unsigned long long)(uintptr_t)gptr;
  u32x4 g0;
  g0.x = 1u;                                  // count=1, user descriptor
  g0.y = lds_off;                             // lds_addr (bytes)
  g0.z = (unsigned)ga;                        // global_addr[31:0]
  g0.w = (unsigned)(ga >> 32) | (2u << 30);   // global_addr[56:32] | type=2
  i32x8 g1;
  g1[0] = (int)(2u << 16);                    // data_size = 2 (4 bytes)
  g1[1] = (int)((tile_w & 0xFFFFu) << 16);    // tensor_dim0[15:0] (= tile_w: no OOB)
  g1[2] = (int)((tile_w >> 16) | ((tile_h & 0xFFFFu) << 16)); // td0 hi | td1 lo
  g1[3] = (int)((tile_h >> 16) | (tile_w << 16));             // td1 hi | tile_dim0
  g1[4] = (int)tile_h;                        // tile_dim1 (tile_dim2 = 0)
  g1[5] = (int)row_stride_elems;              // tensor_dim0_stride[31:0]
  g1[6] = 0;                                  // stride0 hi | stride1 lo
  g1[7] = 0;                                  // stride1 hi
  i32x4 z4 = {0, 0, 0, 0};
#if __has_include(<hip/amd_detail/amd_gfx1250_TDM.h>)
  i32x8 z8 = {0, 0, 0, 0, 0, 0, 0, 0};
  __builtin_amdgcn_tensor_load_to_lds(g0, g1, z4, z4, z8, 0);  // clang-23 / therock
#else
  __builtin_amdgcn_tensor_load_to_lds(g0, g1, z4, z4, 0);      // ROCm 7.2 clang-22
#endif
}
#endif

// MODE 0: X = img           ; skel  = relu(X - open(X))            (no img_out)
// MODE 1: X = erode(img_in) ; img_out = X; skel += relu(d - skel*d), d = relu(X - open(X))
template <int MODE>
__global__ __launch_bounds__(256) void soft_skel_step(
    const float* __restrict__ img_in, float* __restrict__ img_out,
    float* __restrict__ skel, int H, int W) {
  __shared__ float sA[LW * LW];   // img tile, then reused for ero2
  __shared__ float sB[EW * LW];   // E tile (stride LW)

  const int tid = (int)threadIdx.x;
  const int gx0 = (int)blockIdx.x * TILE;
  const int gy0 = (int)blockIdx.y * TILE;
  const size_t plane = (size_t)H * W;
  const float* in = img_in + (size_t)blockIdx.z * plane;

  const bool interior = (gx0 >= HALO) && (gx0 + TILE + HALO <= W) &&
                        (gy0 >= HALO) && (gy0 + TILE + HALO <= H);
  const float PINF = __builtin_huge_valf();

  // ---- Stage 0: load 38x38 input tile into LDS ----
  if (interior) {
#if HAVE_TDM
    if (tid < 32) {  // wave 0 issues the tile DMA once
      unsigned lds_off = (unsigned)(uintptr_t)(&sA[0]);  // low 32 bits = LDS byte offset
      const float* src = in + (size_t)(gy0 - HALO) * W + (gx0 - HALO);
      tdm_load_2d(lds_off, src, LW, LW, (unsigned)W);
    }
    __builtin_amdgcn_s_wait_tensorcnt(0);
#else
    for (int i = tid; i < LW * LW; i += 256) {
      int r = i / LW, c = i % LW;
      sA[i] = in[(size_t)(gy0 - HALO + r) * W + (gx0 - HALO + c)];
    }
#endif
  } else {
    // border tiles: guarded loads, +inf fill (min-pool padding)
    for (int i = tid; i < LW * LW; i += 256) {
      int r = i / LW, c = i % LW;
      int y = gy0 - HALO + r, x = gx0 - HALO + c;
      sA[i] = (y >= 0 && y < H && x >= 0 && x < W) ? in[(size_t)y * W + x] : PINF;
    }
  }
  __syncthreads();

  // ---- Stage 1: sB = (MODE ? erode(img) : img) over 36x36 (halo 2) ----
  // Out-of-image positions must hold +inf (pad for the next erode).
  for (int i = tid; i < EW * EW; i += 256) {
    int r = i / EW, c = i % EW;
    float v;
    if (MODE == 0) {
      v = sA[(r + 1) * LW + (c + 1)];          // already +inf outside image
    } else {
      int y = gy0 - 2 + r, x = gx0 - 2 + c;
      if (!interior && (y < 0 || y >= H || x < 0 || x >= W)) {
        v = PINF;
      } else {
        int a = (r + 1) * LW + (c + 1);
        float ctr = sA[a];
        float vv = fminf(fminf(sA[a - LW], ctr), sA[a + LW]);
        float hh = fminf(fminf(sA[a - 1], ctr), sA[a + 1]);
        v = fminf(vv, hh);                      // 5-point cross min
      }
    }
    sB[r * LW + c] = v;
  }
  __syncthreads();

  // ---- Stage 2: sA = erode(sB) over 34x34 (halo 1) ----
  // Out-of-image positions must hold -inf (pad for the dilate).
  for (int i = tid; i < DW * DW; i += 256) {
    int r = i / DW, c = i % DW;
    int y = gy0 - 1 + r, x = gx0 - 1 + c;
    float v;
    if (!interior && (y < 0 || y >= H || x < 0 || x >= W)) {
      v = -PINF;
    } else {
      int b = (r + 1) * LW + (c + 1);
      float ctr = sB[b];
      float vv = fminf(fminf(sB[b - LW], ctr), sB[b + LW]);
      float hh = fminf(fminf(sB[b - 1], ctr), sB[b + 1]);
      v = fminf(vv, hh);
    }
    sA[r * LW + c] = v;
  }
  __syncthreads();

  // ---- Stage 3: open = 3x3 max(sA); delta; write skel (and img for MODE 1) ----
  float* so = skel + (size_t)blockIdx.z * plane;
  float* io = img_out + (size_t)blockIdx.z * plane;
  for (int i = tid; i < TILE * TILE; i += 256) {
    int r = i / TILE, c = i % TILE;
    int a = (r + 1) * LW + (c + 1);
    float m0 = fmaxf(fmaxf(sA[a - LW - 1], sA[a - LW]), sA[a - LW + 1]);
    float m1 = fmaxf(fmaxf(sA[a - 1],      sA[a]),      sA[a + 1]);
    float m2 = fmaxf(fmaxf(sA[a + LW - 1], sA[a + LW]), sA[a + LW + 1]);
    float open = fmaxf(fmaxf(m0, m1), m2);
    float e = sB[(r + 2) * LW + (c + 2)];
    float delta = fmaxf(e - open, 0.0f);
    size_t idx = (size_t)(gy0 + r) * W + (gx0 + c);
    if (MODE == 0) {
      so[idx] = delta;                          // full overwrite: deterministic start
    } else {
      float s = so[idx];
      so[idx] = s + fmaxf(delta - s * delta, 0.0f);
      io[idx] = e;                              // img advances by one erode
    }
  }
}

extern "C" void kernel_launch(void* const* d_in, const int* in_sizes, int n_in,
                              void* d_out, int out_size, void* d_ws, size_t ws_size,
                              hipStream_t stream) {
  (void)n_in; (void)out_size; (void)ws_size;
  const int W = 1024, H = 1024;                 // reference: (4,1,1024,1024) fp32
  const float* img = (const float*)d_in[0];
  const int total = in_sizes[0];
  const int N = total / (W * H);
  float* skel = (float*)d_out;                  // skel accumulates in place in d_out
  float* buf0 = (float*)d_ws;                   // needs 2*total*4 bytes of workspace
  float* buf1 = buf0 + (size_t)total;

  dim3 grid(W / TILE, H / TILE, N), block(256);

  // skel = relu(img - open(img)); img unchanged (d_in never mutated)
  soft_skel_step<0><<<grid, block, 0, stream>>>(img, buf0, skel, H, W);

  // 40 scan steps: img = erode(img); skel += relu(d - skel*d)
  const float* cur = img;
  float* nxt = buf0;
  for (int k = 0; k < 40; ++k) {
    soft_skel_step<1><<<grid, block, 0, stream>>>(cur, nxt, skel, H, W);
    cur = nxt;
    nxt = (nxt == buf0) ? buf1 : buf0;
  }
}